// MultiplePointExtractor_74156905332945
// MI455X (gfx1250) — compile-verified
//
#include <hip/hip_runtime.h>

// MultiplePointExtractor for MI455X (gfx1250, wave32).
// Gather-bandwidth kernel (~822 MB L2-resident reads + 205 MB streamed
// writes). CDNA5 path: per-lane gathers are software-pipelined 2 stages deep
// through the async engine (global_load_async_to_lds_b32 + ASYNCcnt), taps
// are consumed from LDS via ds_load_b128, and the output is streamed with
// non-temporal stores so the feature pyramids (~183 MB) stay resident in the
// 192 MB L2.

#define NROI 1024
#define NP   49
#define NC   256
#define NLVL 4

// Async gather: copies 4 bytes from global memory at sbase plus byte offset goff to the LDS address in ldsA, tracked by ASYNCcnt.
#define ASYNC_TAP(ldsA, goff, sbase)                                     \
    asm volatile("global_load_async_to_lds_b32 %0, %1, %2"               \
                 :: "v"(ldsA), "v"(goff), "s"(sbase) : "memory")

#define WAIT_ASYNCCNT(n) \
    asm volatile("s_wait_asynccnt %0" :: "i"(n) : "memory")

#define WAIT_DSCNT0() \
    asm volatile("s_wait_dscnt 0x0" ::: "memory")

__global__ __launch_bounds__(256)
void MultiplePointExtractor_74156905332945_kernel(
    const float* __restrict__ f0, const float* __restrict__ f1,
    const float* __restrict__ f2, const float* __restrict__ f3,
    const float* __restrict__ rois, const float* __restrict__ points,
    float* __restrict__ out)
{
    __shared__ int   sOff[NLVL][NP][4];
    __shared__ float sW[NLVL][NP][4];
    __shared__ __align__(16) float sTap[2][256][4];   // double-buffered taps
    __shared__ int   sBatch;

    const int n = blockIdx.x;     // roi index
    const int t = threadIdx.x;    // 0..255 (8 waves)

    const int   Hs[NLVL]      = {200, 100, 50, 25};
    const int   Ws[NLVL]      = {336, 168, 84, 42};
    const float strides[NLVL] = {4.f, 8.f, 16.f, 32.f};

    if (t == 0) sBatch = (int)rois[(size_t)n * 5 + 0];

    // ---- Phase 1: per-(level, point) tap offsets & weights into LDS ----
    if (t < NLVL * NP) {
        const int L = t / NP;
        const int p = t - L * NP;

        const float x1r = rois[(size_t)n * 5 + 1];
        const float y1r = rois[(size_t)n * 5 + 2];
        const float x2r = rois[(size_t)n * 5 + 3];
        const float y2r = rois[(size_t)n * 5 + 4];
        const float px  = points[((size_t)n * NP + p) * 2 + 0];
        const float py  = points[((size_t)n * NP + p) * 2 + 1];

        const float cx = x1r + px * (x2r - x1r);   // image coords
        const float cy = y1r + py * (y2r - y1r);

        const int   W = Ws[L], H = Hs[L];
        const float s = strides[L];

        // Match reference arithmetic exactly (align_corners=False).
        const float gx = 2.0f * (cx / ((float)W * s)) - 1.0f;
        const float gy = 2.0f * (cy / ((float)H * s)) - 1.0f;
        const float x  = ((gx + 1.0f) * (float)W - 1.0f) * 0.5f;
        const float y  = ((gy + 1.0f) * (float)H - 1.0f) * 0.5f;

        const float x0f = floorf(x), y0f = floorf(y);
        const float wx1 = x - x0f,  wx0 = 1.0f - wx1;
        const float wy1 = y - y0f,  wy0 = 1.0f - wy1;
        const int x0 = (int)x0f, y0 = (int)y0f;
        const int x1 = x0 + 1,   y1 = y0 + 1;

        // Validity mask is separable: fold it into per-axis weights.
        const float ax0 = (x0 >= 0 && x0 < W) ? wx0 : 0.0f;
        const float ax1 = (x1 >= 0 && x1 < W) ? wx1 : 0.0f;
        const float ay0 = (y0 >= 0 && y0 < H) ? wy0 : 0.0f;
        const float ay1 = (y1 >= 0 && y1 < H) ? wy1 : 0.0f;

        const int x0c = x0 < 0 ? 0 : (x0 > W - 1 ? W - 1 : x0);
        const int x1c = x1 < 0 ? 0 : (x1 > W - 1 ? W - 1 : x1);
        const int y0c = y0 < 0 ? 0 : (y0 > H - 1 ? H - 1 : y0);
        const int y1c = y1 < 0 ? 0 : (y1 > H - 1 ? H - 1 : y1);

        sOff[L][p][0] = y0c * W + x0c;
        sOff[L][p][1] = y0c * W + x1c;
        sOff[L][p][2] = y1c * W + x0c;
        sOff[L][p][3] = y1c * W + x1c;
        sW[L][p][0] = ax0 * ay0;
        sW[L][p][1] = ax1 * ay0;
        sW[L][p][2] = ax0 * ay1;
        sW[L][p][3] = ax1 * ay1;
    }
    __syncthreads();

    // ---- Phase 2: async-pipelined gather + blend.
    // lane -> point (stores coalesce over the P-contiguous output),
    // thread group -> channel phase; 64 channel stages per level. ----
    const int p    = t & 63;    // point slot (0..63; 49..63 idle)
    const int cgrp = t >> 6;    // channel phase 0..3 (wave-uniform)
    if (p >= NP) return;

    const int bi = sBatch;
    const float* const feats[NLVL] = {f0, f1, f2, f3};

    // Low 32 bits of a generic LDS pointer == wave-relative LDS byte address.
    const unsigned ldsT = (unsigned)(uintptr_t)&sTap[0][t][0];
    const float4* __restrict__ buf0 = (const float4*)&sTap[0][t][0];
    const float4* __restrict__ buf1 = (const float4*)&sTap[1][t][0];

    for (int L = 0; L < NLVL; ++L) {
        const int HW = Hs[L] * Ws[L];
        const float* __restrict__ base = feats[L] + (size_t)bi * NC * HW;

        const unsigned g0 = (unsigned)sOff[L][p][0] * 4u;
        const unsigned g1 = (unsigned)sOff[L][p][1] * 4u;
        const unsigned g2 = (unsigned)sOff[L][p][2] * 4u;
        const unsigned g3 = (unsigned)sOff[L][p][3] * 4u;
        const float w0 = sW[L][p][0], w1 = sW[L][p][1];
        const float w2 = sW[L][p][2], w3 = sW[L][p][3];

        float* __restrict__ outL =
            out + ((size_t)n * (NLVL * NC) + (size_t)L * NC) * NP + p;

        const size_t planeStride = (size_t)4 * HW;         // c advances by 4
        const float* plane = base + (size_t)cgrp * HW;     // stage 0 plane

        // Prologue: stages 0 (buf0) and 1 (buf1).
        ASYNC_TAP(ldsT + 0u,    g0, plane);
        ASYNC_TAP(ldsT + 4u,    g1, plane);
        ASYNC_TAP(ldsT + 8u,    g2, plane);
        ASYNC_TAP(ldsT + 12u,   g3, plane);
        const float* plane1 = plane + planeStride;
        ASYNC_TAP(ldsT + 4096u + 0u,  g0, plane1);
        ASYNC_TAP(ldsT + 4096u + 4u,  g1, plane1);
        ASYNC_TAP(ldsT + 4096u + 8u,  g2, plane1);
        ASYNC_TAP(ldsT + 4096u + 12u, g3, plane1);

        const float* planeNext = plane1 + planeStride;     // stage 2 plane

        // Steady state: stages 0..61 (manually 2x unrolled: even->buf0,
        // odd->buf1), each consumes the oldest stage and issues stage s+2.
        for (int s = 0; s < 62; s += 2) {
            {   // even stage s -> buf0
                WAIT_ASYNCCNT(4);
                const float4 v = *buf0;
                WAIT_DSCNT0();                 // WAR: taps in VGPRs before reuse
                ASYNC_TAP(ldsT + 0u,  g0, planeNext);
                ASYNC_TAP(ldsT + 4u,  g1, planeNext);
                ASYNC_TAP(ldsT + 8u,  g2, planeNext);
                ASYNC_TAP(ldsT + 12u, g3, planeNext);
                planeNext += planeStride;
                const float acc =
                    fmaf(w3, v.w, fmaf(w2, v.z, fmaf(w1, v.y, w0 * v.x)));
                __builtin_nontemporal_store(
                    acc, outL + (size_t)(cgrp + 4 * s) * NP);
            }
            {   // odd stage s+1 -> buf1
                WAIT_ASYNCCNT(4);
                const float4 v = *buf1;
                WAIT_DSCNT0();
                ASYNC_TAP(ldsT + 4096u + 0u,  g0, planeNext);
                ASYNC_TAP(ldsT + 4096u + 4u,  g1, planeNext);
                ASYNC_TAP(ldsT + 4096u + 8u,  g2, planeNext);
                ASYNC_TAP(ldsT + 4096u + 12u, g3, planeNext);
                planeNext += planeStride;
                const float acc =
                    fmaf(w3, v.w, fmaf(w2, v.z, fmaf(w1, v.y, w0 * v.x)));
                __builtin_nontemporal_store(
                    acc, outL + (size_t)(cgrp + 4 * (s + 1)) * NP);
            }
        }

        // Drain: stages 62 (buf0) and 63 (buf1).
        WAIT_ASYNCCNT(0);
        {
            const float4 v = *buf0;
            const float acc =
                fmaf(w3, v.w, fmaf(w2, v.z, fmaf(w1, v.y, w0 * v.x)));
            __builtin_nontemporal_store(
                acc, outL + (size_t)(cgrp + 4 * 62) * NP);
        }
        {
            const float4 v = *buf1;
            const float acc =
                fmaf(w3, v.w, fmaf(w2, v.z, fmaf(w1, v.y, w0 * v.x)));
            __builtin_nontemporal_store(
                acc, outL + (size_t)(cgrp + 4 * 63) * NP);
        }
    }
}

extern "C" void kernel_launch(void* const* d_in, const int* in_sizes, int n_in,
                              void* d_out, int out_size, void* d_ws, size_t ws_size,
                              hipStream_t stream) {
    const float* f0     = (const float*)d_in[0];
    const float* f1     = (const float*)d_in[1];
    const float* f2     = (const float*)d_in[2];
    const float* f3     = (const float*)d_in[3];
    const float* rois   = (const float*)d_in[4];
    const float* points = (const float*)d_in[5];
    float* out          = (float*)d_out;

    MultiplePointExtractor_74156905332945_kernel<<<NROI, 256, 0, stream>>>(
        f0, f1, f2, f3, rois, points, out);
}